// PiecewiseLinearExpansion2d_14731737825613
// MI455X (gfx1250) — compile-verified
//
#include <hip/hip_runtime.h>
#include <hip/hip_bf16.h>

#define HW_   16384   // H*W = 128*128
#define C_    32
#define P_    8
#define TPB   128     // threads per block
#define PXT   4       // pixels per thread (float4)
#define NPXB  (TPB * PXT)  // 512 pixels per block

// Native clang vector types (HIP's float4 is a class -> rejected by builtins)
typedef float vf4 __attribute__((ext_vector_type(4)));
typedef int   vi4 __attribute__((vector_size(16)));

// Address-space-qualified pointee types for the async global->LDS builtin
typedef __attribute__((address_space(1))) vi4 as1_vi4;  // global (__device__)
typedef __attribute__((address_space(3))) vi4 as3_vi4;  // LDS (__shared__)

#if defined(__has_builtin)
#if __has_builtin(__builtin_amdgcn_global_load_async_to_lds_b128) && \
    __has_builtin(__builtin_amdgcn_s_wait_asynccnt)
#define HAVE_ASYNC_LDS 1
#endif
#endif

__global__ __launch_bounds__(TPB)
void PiecewiseLinearExpansion2d_kernel(const float* __restrict__ x,
                                       const float* __restrict__ pos_g,
                                       float* __restrict__ out)
{
    __shared__ float lds_x[C_ * NPXB];   // 64 KB tile: [channel][pixel-in-block]

    const int tid       = threadIdx.x;
    const int block_px0 = blockIdx.x * NPXB;      // global pixel of block start
    const int b         = block_px0 / HW_;        // batch image (block never spans two)
    const int hw0       = block_px0 % HW_;
    const int px        = tid * PXT;              // pixel offset inside block

    const float* xb = x + (size_t)b * C_ * HW_ + hw0 + px;

    // ---- Stage x tile into LDS (CDNA5 async global->LDS DMA, ASYNCcnt) ----
#if HAVE_ASYNC_LDS
    #pragma unroll
    for (int c = 0; c < C_; ++c) {
        __builtin_amdgcn_global_load_async_to_lds_b128(
            (as1_vi4*)(xb + c * HW_),
            (as3_vi4*)(&lds_x[c * NPXB + px]),
            /*offset=*/0, /*cpol=*/0);
    }
#else
    #pragma unroll
    for (int c = 0; c < C_; ++c) {
        *(vf4*)(&lds_x[c * NPXB + px]) = *(const vf4*)(xb + c * HW_);
    }
#endif

    // ---- positions + reciprocal spacings (overlaps async latency) ----
    float pos[P_];
    #pragma unroll
    for (int k = 0; k < P_; ++k) pos[k] = pos_g[k];
    float inv[P_ - 1];
    #pragma unroll
    for (int k = 0; k < P_ - 1; ++k) inv[k] = 1.0f / (pos[k + 1] - pos[k]);

#if HAVE_ASYNC_LDS
    __builtin_amdgcn_s_wait_asynccnt(0);
#endif

    // ---- Pass 1: channel mean of clamped x for this thread's 4 pixels ----
    float sum[PXT] = {0.f, 0.f, 0.f, 0.f};
    for (int c = 0; c < C_; ++c) {
        const vf4 v = *(const vf4*)(&lds_x[c * NPXB + px]);
        #pragma unroll
        for (int j = 0; j < PXT; ++j) {
            sum[j] += fminf(fmaxf(v[j], pos[0]), pos[P_ - 1]);
        }
    }

    // ---- Basis values (8 per pixel) ----
    float vals[P_][PXT];
    #pragma unroll
    for (int j = 0; j < PXT; ++j) {
        const float m = sum[j] * (1.0f / (float)C_);
        vals[0][j]      = (m <= pos[1]) ? (pos[1] - m) * inv[0] : 0.0f;
        vals[P_ - 1][j] = (m >= pos[P_ - 2]) ? (m - pos[P_ - 2]) * inv[P_ - 2] : 0.0f;
        #pragma unroll
        for (int i = 1; i < P_ - 1; ++i) {
            const float l = pos[i - 1], cm = pos[i], r = pos[i + 1];
            float v = 0.0f;
            if (m >= l && m <= cm) v  = (m - l) * inv[i - 1];
            if (m > cm && m <= r)  v += (r - m) * inv[i];
            vals[i][j] = v;
        }
    }

    // ---- Pass 2: expand & stream out (write-once -> non-temporal) ----
    float* ob = out + (size_t)b * (C_ * P_) * HW_ + hw0 + px;
    for (int c = 0; c < C_; ++c) {
        const vf4 v = *(const vf4*)(&lds_x[c * NPXB + px]);
        float* oc = ob + (size_t)(c * P_) * HW_;
        #pragma unroll
        for (int p = 0; p < P_; ++p) {
            vf4 o;
            o.x = v.x * vals[p][0];
            o.y = v.y * vals[p][1];
            o.z = v.z * vals[p][2];
            o.w = v.w * vals[p][3];
            __builtin_nontemporal_store(o, (vf4*)(oc + (size_t)p * HW_));
        }
    }
}

extern "C" void kernel_launch(void* const* d_in, const int* in_sizes, int n_in,
                              void* d_out, int out_size, void* d_ws, size_t ws_size,
                              hipStream_t stream) {
    const float* x   = (const float*)d_in[0];
    const float* pos = (const float*)d_in[1];
    float* out       = (float*)d_out;

    const int total_px = in_sizes[0] / C_;       // B*H*W = 262144
    const int grid     = total_px / NPXB;        // 512 blocks
    PiecewiseLinearExpansion2d_kernel<<<grid, TPB, 0, stream>>>(x, pos, out);
}